// Target_10333691314263
// MI455X (gfx1250) — compile-verified
//
#include <hip/hip_runtime.h>
#include <math.h>

// FCOS-style target assignment for MI455X (gfx1250, wave32).
//
// Levels: strides {8,16,32,64,128}, fmaps {160,80,40,20,10}^2 -> 34100 pixels.
// Outputs: cls (34100 x 82) followed by regr (34100 x 6), flat f32.
//
// Strategy:
//   * 2560B gt-box table broadcast to LDS once per block via CDNA5 async
//     global->LDS DMA (GLOBAL_LOAD_ASYNC_TO_LDS_B128 + s_wait_asynccnt).
//   * 128 boxes preprocessed (shrunk int bounds, validity, label, sel_w)
//     by the first 128 lanes into LDS arrays.
//   * 1 thread = 1 pixel: argmin over 128 boxes (first-min tie-break to
//     match jnp.argmin), winner decoded once.
//   * Results staged in LDS, then block-coalesced contiguous stores of the
//     82-wide cls rows and 6-wide regr rows (flat index == row*82+col).

#define NUM_CLS   80
#define NBOX      128
#define TOTAL_PIX 34100
#define BIG_AREA  1.0e7f
#define NBLOCKS   135

#ifndef USE_ASYNC_LDS
#if defined(__has_builtin)
#if __has_builtin(__builtin_amdgcn_global_load_async_to_lds_b128)
#define USE_ASYNC_LDS 1
#endif
#if __has_builtin(__builtin_amdgcn_s_wait_asynccnt)
#define HAVE_WAIT_ASYNC_BUILTIN 1
#endif
#endif
#endif
#ifndef USE_ASYNC_LDS
#define USE_ASYNC_LDS 0
#endif
#ifndef HAVE_WAIT_ASYNC_BUILTIN
#define HAVE_WAIT_ASYNC_BUILTIN 0
#endif

// 16-byte vector type matching the async-LDS builtin's pointee type
// (clang reported: "int __vector(4) __device__ *" i.e. v4i addrspace(1)*).
typedef int v4i __attribute__((vector_size(4 * sizeof(int))));
typedef v4i __attribute__((address_space(1))) * gv4i_ptr;   // global
typedef v4i __attribute__((address_space(3))) * lv4i_ptr;   // LDS

__global__ __launch_bounds__(256)
void fcos_target_kernel(const float* __restrict__ gt,
                        const float* __restrict__ fsw,
                        float* __restrict__ out)
{
    __shared__ __align__(16) float sRaw[NBOX * 5];   // raw gt rows
    __shared__ float sX1[NBOX], sY1[NBOX], sX2[NBOX], sY2[NBOX], sSelw[NBOX];
    __shared__ int   sPx1[NBOX], sPy1[NBOX], sPx2[NBOX], sPy2[NBOX];
    __shared__ int   sLab[NBOX], sValid[NBOX];
    __shared__ float sRes[256][6];   // dl,dt,dr,db (scaled+masked), soft, is_pos
    __shared__ int   sResLab[256];   // winner label if positive else -1

    const int tid = threadIdx.x;
    const int blk = blockIdx.x;

    const int strideTab[5] = {8, 16, 32, 64, 128};
    const int fwTab[5]     = {160, 80, 40, 20, 10};
    const int pixBase[5]   = {0, 25600, 32000, 33600, 34000};
    const int blkBase[5]   = {0, 100, 125, 132, 134};

    int lvl = 0;
#pragma unroll
    for (int l = 1; l < 5; ++l) if (blk >= blkBase[l]) lvl = l;

    const int   stride = strideTab[lvl];
    const int   fw     = fwTab[lvl];
    const int   npix   = fw * fw;
    const float fstride = (float)stride;

    // ---- Stage 1: broadcast gt (128x5 f32 = 2560B = 160 x b128) into LDS ----
#if USE_ASYNC_LDS
    if (tid < 160) {
        __builtin_amdgcn_global_load_async_to_lds_b128(
            (gv4i_ptr)(gt + tid * 4),
            (lv4i_ptr)(sRaw + tid * 4),
            0, 0);
    }
#if HAVE_WAIT_ASYNC_BUILTIN
    __builtin_amdgcn_s_wait_asynccnt(0);
#else
    asm volatile("s_wait_asynccnt 0" ::: "memory");
#endif
#else
    if (tid < 160) {
        ((float4*)sRaw)[tid] = ((const float4*)gt)[tid];
    }
#endif
    __syncthreads();

    // ---- Stage 2: per-box preprocessing (first 128 lanes) ----
    if (tid < NBOX) {
        const float x1 = sRaw[tid * 5 + 0];
        const float y1 = sRaw[tid * 5 + 1];
        const float x2 = sRaw[tid * 5 + 2];
        const float y2 = sRaw[tid * 5 + 3];
        const int   lab = (int)sRaw[tid * 5 + 4];
        const int   vld = (fabsf(x1) > 0.f) | (fabsf(y1) > 0.f) |
                          (fabsf(x2) > 0.f) | (fabsf(y2) > 0.f);

        const float inv = 1.0f / fstride;              // stride is pow2: exact
        const float bx1 = x1 * inv, by1 = y1 * inv;
        const float bx2 = x2 * inv, by2 = y2 * inv;
        const float cx = (bx1 + bx2) * 0.5f;
        const float cy = (by1 + by2) * 0.5f;
        const float hw = (bx2 - bx1) * 0.2f * 0.5f;    // SR = 0.2
        const float hh = (by2 - by1) * 0.2f * 0.5f;

        int px1 = (int)fminf(fmaxf(floorf(cx - hw), 0.f), (float)(fw - 1));
        int py1 = (int)fminf(fmaxf(floorf(cy - hh), 0.f), (float)(fw - 1));
        int px2 = (int)ceilf(cx + hw); px2 = min(max(px2, px1 + 1), fw);
        int py2 = (int)ceilf(cy + hh); py2 = min(max(py2, py1 + 1), fw);

        sX1[tid] = x1; sY1[tid] = y1; sX2[tid] = x2; sY2[tid] = y2;
        sPx1[tid] = px1; sPy1[tid] = py1; sPx2[tid] = px2; sPy2[tid] = py2;
        sLab[tid] = lab; sValid[tid] = vld;
        sSelw[tid] = fsw[tid * 5 + lvl];
    }
    __syncthreads();

    // ---- Stage 3: per-pixel argmin over 128 boxes ----
    const int lpBase = (blk - blkBase[lvl]) * 256;
    const int lp     = lpBase + tid;

    if (lp < npix) {
        const int   ix = lp % fw;
        const int   iy = lp / fw;
        const float sx = ((float)ix + 0.5f) * fstride;
        const float sy = ((float)iy + 0.5f) * fstride;

        float best = 3.402823466e+38f;  // > BIG_AREA so b=0 always taken first
        int   win  = 0;
        for (int b = 0; b < NBOX; ++b) {
            const float dl = fmaxf(sx - sX1[b], 0.f);
            const float dt = fmaxf(sy - sY1[b], 0.f);
            const float dr = fmaxf(sX2[b] - sx, 0.f);
            const float db = fmaxf(sY2[b] - sy, 0.f);
            const bool inb = (ix >= sPx1[b]) & (ix < sPx2[b]) &
                             (iy >= sPy1[b]) & (iy < sPy2[b]) & (sValid[b] != 0);
            const float area = inb ? (dl + dr) * (dt + db) : BIG_AREA;
            if (area < best) { best = area; win = b; }   // first-min == jnp.argmin
        }

        // decode winner
        const float dl = fmaxf(sx - sX1[win], 0.f);
        const float dt = fmaxf(sy - sY1[win], 0.f);
        const float dr = fmaxf(sX2[win] - sx, 0.f);
        const float db = fmaxf(sY2[win] - sy, 0.f);
        const bool  inb = (ix >= sPx1[win]) & (ix < sPx2[win]) &
                          (iy >= sPy1[win]) & (iy < sPy2[win]) & (sValid[win] != 0);
        const float is_pos = inb ? 1.0f : 0.0f;

        const float num = fminf(dl, dr) * fminf(dt, db);
        const float den = fmaxf(dl, dr) * fmaxf(dt, db);
        const float ap  = (den > 0.f) ? (num / fmaxf(den, 1e-12f)) : 0.0f;
        const float soft = inb ? (ap * sSelw[win]) : 1.0f;

        const float rs = 1.0f / (4.0f * fstride);       // pow2: exact reciprocal
        sRes[tid][0] = inb ? dl * rs : 0.0f;
        sRes[tid][1] = inb ? dt * rs : 0.0f;
        sRes[tid][2] = inb ? dr * rs : 0.0f;
        sRes[tid][3] = inb ? db * rs : 0.0f;
        sRes[tid][4] = soft;
        sRes[tid][5] = is_pos;
        sResLab[tid] = inb ? sLab[win] : -1;
    }
    __syncthreads();

    // ---- Stage 4: block-coalesced contiguous stores ----
    const int nvalid = min(256, npix - lpBase);
    const int r0     = pixBase[lvl] + lpBase;

    // cls rows: out[(r0+rl)*82 + c] == clsBase[i] with i = rl*82 + c
    {
        float* clsBase = out + (size_t)r0 * (NUM_CLS + 2);
        const int total = nvalid * (NUM_CLS + 2);
        for (int i = tid; i < total; i += 256) {
            const int rl = i / (NUM_CLS + 2);
            const int c  = i - rl * (NUM_CLS + 2);
            float v;
            if (c < NUM_CLS)       v = (c == sResLab[rl]) ? 1.0f : 0.0f;
            else if (c == NUM_CLS) v = sRes[rl][4];   // soft
            else                   v = sRes[rl][5];   // is_pos
            clsBase[i] = v;
        }
    }
    // regr rows: 6 wide, placed after full cls tensor
    {
        float* rgBase = out + (size_t)TOTAL_PIX * (NUM_CLS + 2) + (size_t)r0 * 6;
        const int total = nvalid * 6;
        for (int i = tid; i < total; i += 256) {
            const int rl = i / 6;
            const int c  = i - rl * 6;
            rgBase[i] = sRes[rl][c];   // [dl,dt,dr,db,soft,is_pos]
        }
    }
}

extern "C" void kernel_launch(void* const* d_in, const int* in_sizes, int n_in,
                              void* d_out, int out_size, void* d_ws, size_t ws_size,
                              hipStream_t stream) {
    const float* gt  = (const float*)d_in[0];   // (128,5) f32
    const float* fsw = (const float*)d_in[1];   // (128,5) f32
    // d_in[2] = feature_maps_shape (unused; levels are compile-time constants)
    float* out = (float*)d_out;                 // 34100*82 cls then 34100*6 regr

    fcos_target_kernel<<<dim3(NBLOCKS), dim3(256), 0, stream>>>(gt, fsw, out);
}